// GNNSingleLayer_79422535238247
// MI455X (gfx1250) — compile-verified
//
#include <hip/hip_runtime.h>
#include <hip/hip_bf16.h>

// GNN single layer: GCNConv (x=hW, sym-norm scatter-add) + PReLU + LayerNorm.
// fp32 throughout (reference is fp32); GEMM uses V_WMMA_F32_16X16X4_F32.

typedef __attribute__((ext_vector_type(2))) float v2f;
typedef __attribute__((ext_vector_type(8))) float v8f;

#define D_FEAT 128
// Frag-ready packed W layout in LDS: Wp[(k>>1)*WP_STRIDE + col*2 + (k&1)].
// Row stride 288 words (mod 64 = 32 banks): the two half-wave k-pair rows
// (kk>>1 and kk>>1 + 1) read disjoint 32-bank windows -> conflict-free b64.
#define WP_STRIDE 288

// ---------------------------------------------------------------------------
// 0) init: zero the output accumulator, set degree count to 1 (self loop)
// ---------------------------------------------------------------------------
__global__ __launch_bounds__(256) void k_init(float* __restrict__ out,
                                              unsigned* __restrict__ cnt,
                                              int N) {
  size_t idx = (size_t)blockIdx.x * 256 + threadIdx.x;
  size_t total = (size_t)N * D_FEAT;
  if (idx < total) out[idx] = 0.0f;
  if (idx < (size_t)N) cnt[idx] = 1u;
}

// ---------------------------------------------------------------------------
// 1) in-degree count over edges (dst side, matching reference segment_sum)
// ---------------------------------------------------------------------------
__global__ __launch_bounds__(256) void k_count(const int* __restrict__ ei,
                                               unsigned* __restrict__ cnt,
                                               int E) {
  int e = blockIdx.x * 256 + threadIdx.x;
  if (e < E) {
    int dst = ei[E + e];  // edge_index row 1
    atomicAdd(&cnt[dst], 1u);
  }
}

// ---------------------------------------------------------------------------
// 2) dis = deg^{-1/2} (deg >= 1 always, self loops included)
// ---------------------------------------------------------------------------
__global__ __launch_bounds__(256) void k_dis(const unsigned* __restrict__ cnt,
                                             float* __restrict__ dis, int N) {
  int i = blockIdx.x * 256 + threadIdx.x;
  if (i < N) dis[i] = rsqrtf((float)cnt[i]);
}

// ---------------------------------------------------------------------------
// 3) x = h_in @ W via V_WMMA_F32_16X16X4_F32.
//    Block: 256 threads = 8 waves; each wave owns a 16-row stripe, block
//    covers 128 rows x 128 cols. W staged once in LDS in frag-ready packed
//    layout so each B frag is one aligned ds_load_b64 into an even VGPR pair.
//    Per wave: 32 K-steps x 8 N-tiles = 256 WMMA ops.
// ---------------------------------------------------------------------------
__global__ __launch_bounds__(256) void k_gemm(const float* __restrict__ h,
                                              const float* __restrict__ W,
                                              float* __restrict__ x, int N) {
  __shared__ float Wlds[(D_FEAT / 2) * WP_STRIDE];  // 64 k-pair rows, 72 KB

  const int t = threadIdx.x;
  // cooperative W stage into packed layout: element (k, col) ->
  // Wlds[(k>>1)*WP_STRIDE + col*2 + (k&1)]
  {
    const float4* Wg = (const float4*)W;
#pragma unroll
    for (int i = 0; i < 16; ++i) {
      int idx = t + 256 * i;          // float4 index, 0..4095
      int k   = idx >> 5;             // 32 float4 per source row
      int c0  = (idx & 31) * 4;
      float4 v = Wg[idx];
      float* dstp = &Wlds[(k >> 1) * WP_STRIDE + c0 * 2 + (k & 1)];
      dstp[0] = v.x; dstp[2] = v.y; dstp[4] = v.z; dstp[6] = v.w;
    }
  }
  __syncthreads();

  const int wave = t >> 5;
  const int lane = t & 31;
  const int r = lane & 15;       // col within 16-wide tile / row-in-half
  const int hi = lane >> 4;      // lane half
  const int klo = hi * 2;        // K sub-row pair for this half (ISA A layout)

  const int m0 = blockIdx.x * 128 + wave * 16;
  int arow_i = m0 + r;
  if (arow_i >= N) arow_i = N - 1;          // clamp loads; stores are guarded
  const float* __restrict__ arow = h + (size_t)arow_i * D_FEAT;

  v8f acc[8] = {};  // 8 N-tiles of 16x16 fp32

  for (int kk = 0; kk < D_FEAT; kk += 4) {
    // A frag (16x4 fp32): lane half selects K {0,1} vs {2,3}; one b64 load
    v2f a = *(const v2f*)(arow + kk + klo);
    // B frags: k-pair row for this half; col = j*16 + r -> word col*2
    const float* bbase = &Wlds[((kk >> 1) + hi) * WP_STRIDE + r * 2];
#pragma unroll
    for (int j = 0; j < 8; ++j) {
      v2f b = *(const v2f*)(bbase + j * 32);
      acc[j] = __builtin_amdgcn_wmma_f32_16x16x4_f32(
          /*neg_a=*/false, a, /*neg_b=*/false, b,
          /*c_mod=*/(short)0, acc[j], /*reuse_a=*/false, /*reuse_b=*/false);
    }
  }

  // C/D layout: VGPR v -> M = v (lanes 0-15) or M = 8+v (lanes 16-31), N = r
#pragma unroll
  for (int v = 0; v < 8; ++v) {
    int orow = m0 + hi * 8 + v;
    if (orow < N) {
      float* xr = x + (size_t)orow * D_FEAT + r;
#pragma unroll
      for (int j = 0; j < 8; ++j) xr[j * 16] = acc[j][v];
    }
  }
}

// ---------------------------------------------------------------------------
// 4) edge scatter: one wave per edge, 4 floats per lane.
//    out[dst] += dis[src]*dis[dst] * x[src]
//    Native no-return fp32 atomics via inline asm (STOREcnt-tracked); the
//    51 MB x and out buffers are L2-resident (192 MB L2).
// ---------------------------------------------------------------------------
__device__ __forceinline__ void atomic_add_f32_native(float* p, float v) {
  asm volatile("global_atomic_add_f32 %0, %1, off"
               :
               : "v"(p), "v"(v)
               : "memory");
}

__global__ __launch_bounds__(256) void k_scatter(const int* __restrict__ ei,
                                                 const float* __restrict__ x,
                                                 const float* __restrict__ dis,
                                                 float* __restrict__ out,
                                                 int E) {
  int e = blockIdx.x * 8 + (threadIdx.x >> 5);
  if (e >= E) return;                       // wave-uniform exit
  int lane = threadIdx.x & 31;
  int src = ei[e];
  int dst = ei[E + e];
  float nrm = dis[src] * dis[dst];
  float4 v = ((const float4*)(x + (size_t)src * D_FEAT))[lane];
  float* o = out + (size_t)dst * D_FEAT + lane * 4;
  atomic_add_f32_native(o + 0, nrm * v.x);
  atomic_add_f32_native(o + 1, nrm * v.y);
  atomic_add_f32_native(o + 2, nrm * v.z);
  atomic_add_f32_native(o + 3, nrm * v.w);
}

// ---------------------------------------------------------------------------
// 5) finalize: h = edge_sum + dis[i]^2 * x[i] + b ; PReLU ; LayerNorm.
//    One wave per row (128 = 32 lanes x float4), wave32 shuffle reduction.
// ---------------------------------------------------------------------------
__global__ __launch_bounds__(256) void k_final(float* __restrict__ out,
                                               const float* __restrict__ x,
                                               const float* __restrict__ dis,
                                               const float* __restrict__ bias,
                                               const float* __restrict__ prelu_a,
                                               const float* __restrict__ gamma,
                                               const float* __restrict__ beta,
                                               int N) {
  int i = blockIdx.x * 8 + (threadIdx.x >> 5);
  if (i >= N) return;                       // wave-uniform exit
  int lane = threadIdx.x & 31;

  float a = prelu_a[0];
  float d = dis[i];
  float selfn = d * d;

  float4 acc = ((const float4*)(out + (size_t)i * D_FEAT))[lane];
  float4 xs  = ((const float4*)(x   + (size_t)i * D_FEAT))[lane];
  float4 bb  = ((const float4*)bias)[lane];

  float h0 = acc.x + selfn * xs.x + bb.x;
  float h1 = acc.y + selfn * xs.y + bb.y;
  float h2 = acc.z + selfn * xs.z + bb.z;
  float h3 = acc.w + selfn * xs.w + bb.w;

  h0 = h0 >= 0.0f ? h0 : a * h0;
  h1 = h1 >= 0.0f ? h1 : a * h1;
  h2 = h2 >= 0.0f ? h2 : a * h2;
  h3 = h3 >= 0.0f ? h3 : a * h3;

  float s  = h0 + h1 + h2 + h3;
  float sq = h0 * h0 + h1 * h1 + h2 * h2 + h3 * h3;
#pragma unroll
  for (int m = 16; m >= 1; m >>= 1) {
    s  += __shfl_xor(s,  m, 32);
    sq += __shfl_xor(sq, m, 32);
  }
  float mu   = s * (1.0f / 128.0f);
  float var  = sq * (1.0f / 128.0f) - mu * mu;
  float rstd = rsqrtf(var + 1e-5f);

  float4 g  = ((const float4*)gamma)[lane];
  float4 be = ((const float4*)beta)[lane];
  float4 o;
  o.x = (h0 - mu) * rstd * g.x + be.x;
  o.y = (h1 - mu) * rstd * g.y + be.y;
  o.z = (h2 - mu) * rstd * g.z + be.z;
  o.w = (h3 - mu) * rstd * g.w + be.w;
  ((float4*)(out + (size_t)i * D_FEAT))[lane] = o;
}

// ---------------------------------------------------------------------------
// launch
// ---------------------------------------------------------------------------
extern "C" void kernel_launch(void* const* d_in, const int* in_sizes, int n_in,
                              void* d_out, int out_size, void* d_ws, size_t ws_size,
                              hipStream_t stream) {
  const float* h_in   = (const float*)d_in[0];
  const int*   ei     = (const int*)d_in[1];   // edge_index (2,E), int32 on device
  const float* W      = (const float*)d_in[2];
  const float* bias   = (const float*)d_in[3];
  const float* pa     = (const float*)d_in[4];
  const float* gamma  = (const float*)d_in[5];
  const float* beta   = (const float*)d_in[6];

  const int N = in_sizes[0] / D_FEAT;
  const int E = in_sizes[1] / 2;

  float*    out = (float*)d_out;
  float*    ws  = (float*)d_ws;
  float*    x   = ws;                                   // N*128 floats
  float*    dis = ws + (size_t)N * D_FEAT;              // N floats
  unsigned* cnt = (unsigned*)(dis + N);                 // N u32

  size_t total = (size_t)N * D_FEAT;
  k_init<<<(unsigned)((total + 255) / 256), 256, 0, stream>>>(out, cnt, N);
  k_count<<<(E + 255) / 256, 256, 0, stream>>>(ei, cnt, E);
  k_dis<<<(N + 255) / 256, 256, 0, stream>>>(cnt, dis, N);
  k_gemm<<<(N + 127) / 128, 256, 0, stream>>>(h_in, W, x, N);
  k_scatter<<<(E + 7) / 8, 256, 0, stream>>>(ei, x, dis, out, E);
  k_final<<<(N + 7) / 8, 256, 0, stream>>>(out, x, dis, bias, pa, gamma, beta, N);
}